// Attention_11690900979891
// MI455X (gfx1250) — compile-verified
//
#include <hip/hip_runtime.h>

// CDNA5 / gfx1250, wave32. Full-fp32 attention block using V_WMMA_F32_16X16X4_F32.
// Each GEMM wave computes a 16x32 output slab (two 16x16 N-tiles sharing one
// A fragment) -> two independent WMMA accumulation chains (no RAW stalls) and
// half the A-operand traffic.

typedef __attribute__((ext_vector_type(2))) float v2f;
typedef __attribute__((ext_vector_type(8))) float v8f;

#define DIMC 512      // channel dim
#define NTOK 256      // tokens per window (16*16)
#define NTAB 961      // (2*16-1)^2 relative-position table size

__device__ __forceinline__ v8f wmma4(v2f a, v2f b, v8f c) {
    // (neg_a, A, neg_b, B, c_mod, C, reuse_a, reuse_b)
    return __builtin_amdgcn_wmma_f32_16x16x4_f32(false, a, false, b, (short)0, c,
                                                 false, false);
}

// ---------------------------------------------------------------------------
// C[b, m, n] = sum_k W[m,k] * X[b,k,n] + bias[m]
// W: M x 512 row-major (shared over batch), X: 2 x 512 x 256, C: 2 x M x 256
// One wave per 16x32 slab of C (two N-tiles).
// ---------------------------------------------------------------------------
__global__ void gemm_wxb_kernel(const float* __restrict__ W,
                                const float* __restrict__ X,
                                const float* __restrict__ bias,
                                float* __restrict__ C, int M) {
    const int K = DIMC, N = NTOK;
    const int slabsN = N / 32;                    // 8 slabs of two 16-tiles
    const int slabsPerBatch = (M / 16) * slabsN;

    int wave = threadIdx.x >> 5;
    int slab = blockIdx.x * (blockDim.x >> 5) + wave;
    if (slab >= 2 * slabsPerBatch) return;        // wave-uniform guard

    int bb = slab / slabsPerBatch;
    int t  = slab - bb * slabsPerBatch;
    int tm = t / slabsN, tn = t - tm * slabsN;

    const float* Xb = X + (size_t)bb * K * N;
    float*       Cb = C + (size_t)bb * M * N;

    int lane = threadIdx.x & 31;
    int half = lane >> 4, l16 = lane & 15;
    int arow  = tm * 16 + l16;         // A fragment row (m)
    int bcol0 = tn * 32 + l16;         // B fragment col of tile 0
    int bcol1 = bcol0 + 16;            // B fragment col of tile 1

    const float* Wr = W + (size_t)arow * K;

    v8f acc0 = {}, acc1 = {};
    #pragma unroll 4
    for (int k = 0; k < K; k += 4) {
        int k0 = k + 2 * half;
        __builtin_prefetch(Wr + k0 + 32, 0, 0);           // global_prefetch_b8
        v2f a = *(const v2f*)(Wr + k0);                   // contiguous pair
        const float* Xr0 = Xb + (size_t)k0 * N;
        const float* Xr1 = Xr0 + N;
        v2f b0, b1;
        b0.x = Xr0[bcol0]; b0.y = Xr1[bcol0];
        b1.x = Xr0[bcol1]; b1.y = Xr1[bcol1];
        acc0 = wmma4(a, b0, acc0);                        // independent chains
        acc1 = wmma4(a, b1, acc1);
    }
    #pragma unroll
    for (int r = 0; r < 8; ++r) {
        int m = tm * 16 + 8 * half + r;
        float bv = bias[m];
        Cb[(size_t)m * N + bcol0] = acc0[r] + bv;
        Cb[(size_t)m * N + bcol1] = acc1[r] + bv;
    }
}

// ---------------------------------------------------------------------------
// S[b, h, g] = scale * sum_d Q[b,d,h] * K[b,d,g] + rpb[rpi(h,g)]
// Q = QKV rows [0,512), K = rows [512,1024) of the per-batch 1536x256 buffer.
// One wave per 16x32 slab of S.
// ---------------------------------------------------------------------------
__global__ void attn_scores_kernel(const float* __restrict__ QKV,
                                   const float* __restrict__ rpb,
                                   float* __restrict__ S) {
    const int K = DIMC, N = NTOK;
    const float scale = 0.04419417382415922f;     // 512^-0.5
    const int slabsN = N / 32;                    // 8
    const int slabsPerBatch = (N / 16) * slabsN;  // 128

    int wave = threadIdx.x >> 5;
    int slab = blockIdx.x * (blockDim.x >> 5) + wave;
    if (slab >= 2 * slabsPerBatch) return;

    int bb = slab / slabsPerBatch;
    int t  = slab - bb * slabsPerBatch;
    int tm = t / slabsN, tn = t - tm * slabsN;

    const float* Qb = QKV + (size_t)bb * 3 * DIMC * N;
    const float* Kb = Qb + (size_t)DIMC * N;

    int lane = threadIdx.x & 31;
    int half = lane >> 4, l16 = lane & 15;
    int hrow  = tm * 16 + l16;   // A: m index = h (A[m,k] = Q[k,m], strided)
    int gcol0 = tn * 32 + l16;   // B: n index = g (tile 0)
    int gcol1 = gcol0 + 16;

    v8f acc0 = {}, acc1 = {};
    #pragma unroll 4
    for (int k = 0; k < K; k += 4) {
        int k0 = k + 2 * half;
        const float* Qr0 = Qb + (size_t)k0 * N;
        const float* Kr0 = Kb + (size_t)k0 * N;
        __builtin_prefetch(Qr0 + 32 * N + hrow, 0, 0);
        __builtin_prefetch(Kr0 + 32 * N + gcol0, 0, 0);
        v2f a, b0, b1;
        a.x  = Qr0[hrow];      a.y  = Qr0[N + hrow];
        b0.x = Kr0[gcol0];     b0.y = Kr0[N + gcol0];
        b1.x = Kr0[gcol1];     b1.y = Kr0[N + gcol1];
        acc0 = wmma4(a, b0, acc0);
        acc1 = wmma4(a, b1, acc1);
    }

    float* Sb = S + (size_t)bb * N * N;
    #pragma unroll
    for (int r = 0; r < 8; ++r) {
        int h = tm * 16 + 8 * half + r;
        // relative position index: ((h/16-g/16)+(h%16-g%16)) mod 961 (python mod)
        int base = (h >> 4) + (h & 15);
        int s0 = base - ((gcol0 >> 4) + (gcol0 & 15));
        int s1 = base - ((gcol1 >> 4) + (gcol1 & 15));
        int i0 = (s0 < 0) ? (s0 + NTAB) : s0;
        int i1 = (s1 < 0) ? (s1 + NTAB) : s1;
        Sb[(size_t)h * N + gcol0] = acc0[r] * scale + rpb[i0];
        Sb[(size_t)h * N + gcol1] = acc1[r] * scale + rpb[i1];
    }
}

// ---------------------------------------------------------------------------
// Row softmax over 256-wide rows, one wave32 per row (8 elems/lane).
// ---------------------------------------------------------------------------
__global__ void softmax_kernel(float* __restrict__ S) {
    int wave = threadIdx.x >> 5;
    int row  = blockIdx.x * (blockDim.x >> 5) + wave;   // 2*256 rows
    if (row >= 2 * NTOK) return;
    int lane = threadIdx.x & 31;
    float* R = S + (size_t)row * NTOK;

    float v[8];
    float m = -3.402823466e38f;
    #pragma unroll
    for (int i = 0; i < 8; ++i) { v[i] = R[lane + 32 * i]; m = fmaxf(m, v[i]); }
    #pragma unroll
    for (int off = 16; off; off >>= 1) m = fmaxf(m, __shfl_xor(m, off, 32));

    float sum = 0.f;
    #pragma unroll
    for (int i = 0; i < 8; ++i) { v[i] = __expf(v[i] - m); sum += v[i]; }
    #pragma unroll
    for (int off = 16; off; off >>= 1) sum += __shfl_xor(sum, off, 32);

    float inv = 1.0f / sum;
    #pragma unroll
    for (int i = 0; i < 8; ++i) R[lane + 32 * i] = v[i] * inv;
}

// ---------------------------------------------------------------------------
// O[b, d, h] = sum_g V[b, d, g] * P[b, h, g]      (V = QKV rows [1024,1536))
// A[m=d,k=g] = V[d,g] (contiguous), B[k=g,n=h] = P[h,g] (contiguous per lane)
// One wave per 16x32 slab of O.
// ---------------------------------------------------------------------------
__global__ void av_gemm_kernel(const float* __restrict__ QKV,
                               const float* __restrict__ P,
                               float* __restrict__ O) {
    const int M = DIMC, K = NTOK, N = NTOK;
    const int slabsN = N / 32;                    // 8
    const int slabsPerBatch = (M / 16) * slabsN;  // 256

    int wave = threadIdx.x >> 5;
    int slab = blockIdx.x * (blockDim.x >> 5) + wave;
    if (slab >= 2 * slabsPerBatch) return;

    int bb = slab / slabsPerBatch;
    int t  = slab - bb * slabsPerBatch;
    int tm = t / slabsN, tn = t - tm * slabsN;

    const float* Vb = QKV + (size_t)bb * 3 * DIMC * NTOK + (size_t)2 * DIMC * NTOK;
    const float* Pb = P + (size_t)bb * NTOK * NTOK;

    int lane = threadIdx.x & 31;
    int half = lane >> 4, l16 = lane & 15;
    int drow  = tm * 16 + l16;
    int hcol0 = tn * 32 + l16;
    int hcol1 = hcol0 + 16;

    const float* Vr  = Vb + (size_t)drow * NTOK;
    const float* Pr0 = Pb + (size_t)hcol0 * NTOK;
    const float* Pr1 = Pb + (size_t)hcol1 * NTOK;

    v8f acc0 = {}, acc1 = {};
    #pragma unroll 4
    for (int k = 0; k < K; k += 4) {
        int k0 = k + 2 * half;
        __builtin_prefetch(Vr + k0 + 32, 0, 0);
        v2f a  = *(const v2f*)(Vr + k0);
        v2f b0 = *(const v2f*)(Pr0 + k0);
        v2f b1 = *(const v2f*)(Pr1 + k0);
        acc0 = wmma4(a, b0, acc0);
        acc1 = wmma4(a, b1, acc1);
    }
    float* Ob = O + (size_t)bb * DIMC * NTOK;
    #pragma unroll
    for (int r = 0; r < 8; ++r) {
        int d = tm * 16 + 8 * half + r;
        Ob[(size_t)d * NTOK + hcol0] = acc0[r];
        Ob[(size_t)d * NTOK + hcol1] = acc1[r];
    }
}

// ---------------------------------------------------------------------------
extern "C" void kernel_launch(void* const* d_in, const int* in_sizes, int n_in,
                              void* d_out, int out_size, void* d_ws, size_t ws_size,
                              hipStream_t stream) {
    const float* x      = (const float*)d_in[0];   // (2, 512, 256, 1)
    const float* qkv_w  = (const float*)d_in[1];   // (1536, 512)
    const float* qkv_b  = (const float*)d_in[2];   // (1536,)
    const float* proj_w = (const float*)d_in[3];   // (512, 512)
    const float* proj_b = (const float*)d_in[4];   // (512,)
    const float* rpb    = (const float*)d_in[5];   // (961, 1)
    float* out = (float*)d_out;                    // (2, 512, 256, 1)

    float* ws  = (float*)d_ws;
    float* QKV = ws;                                   // 2*1536*256 floats
    float* S   = QKV + (size_t)2 * 3 * DIMC * NTOK;    // 2*256*256 floats
    float* O   = S   + (size_t)2 * NTOK * NTOK;        // 2*512*256 floats

    dim3 blk(256);  // 8 wave32 waves

    // 1) QKV projection: 2 * 96 * 8 = 1536 slabs -> 192 blocks
    gemm_wxb_kernel<<<192, blk, 0, stream>>>(qkv_w, x, qkv_b, QKV, 3 * DIMC);
    // 2) Scores + scale + rel-pos bias: 2 * 128 = 256 slabs -> 32 blocks
    attn_scores_kernel<<<32, blk, 0, stream>>>(QKV, rpb, S);
    // 3) Row softmax: 512 rows, 8 rows/block -> 64 blocks
    softmax_kernel<<<64, blk, 0, stream>>>(S);
    // 4) Attention-weighted values: 2 * 256 = 512 slabs -> 64 blocks
    av_gemm_kernel<<<64, blk, 0, stream>>>(QKV, S, O);
    // 5) Output projection: 2 * 32 * 8 = 512 slabs -> 64 blocks
    gemm_wxb_kernel<<<64, blk, 0, stream>>>(proj_w, O, proj_b, out, DIMC);
}